// MessagePassing_52372831207654
// MI455X (gfx1250) — compile-verified
//
#include <hip/hip_runtime.h>

typedef float v2f __attribute__((ext_vector_type(2)));
typedef float v8f __attribute__((ext_vector_type(8)));

#define N_NUC 100000
#define N_NN  1600000
#define EMB   128
#define MSG   32
#define OUT   128
#define LN_EPS 1e-6f

__device__ __forceinline__ float silu_f(float x) {
    return x / (1.0f + __expf(-x));
}

// ---------------------------------------------------------------------------
// Kernel A: P = X @ W + b     X:[N_NUC,EMB]  W:[EMB,MSG]  P:[N_NUC,MSG]
// One wave per 16-row tile; V_WMMA_F32_16X16X4_F32, K=128 in 32 steps.
// ---------------------------------------------------------------------------
__global__ __launch_bounds__(256) void node_proj_kernel(
    const float* __restrict__ X, const float* __restrict__ W,
    const float* __restrict__ bias, float* __restrict__ P)
{
    const int lane  = threadIdx.x & 31;
    const int wave  = threadIdx.x >> 5;
    const int tile  = blockIdx.x * 8 + wave;
    const int NT    = N_NUC / 16;
    if (tile >= NT) return;                 // wave-uniform: EXEC stays all-1s
    const int row16 = lane & 15;
    const int hi    = lane >> 4;            // which K-half this lane holds
    const int row   = tile * 16 + row16;
    const float* xrow = X + (size_t)row * EMB;

    // A fragments held in registers, reused across both N-tiles
    v2f af[EMB / 4];
#pragma unroll
    for (int k = 0; k < EMB / 4; ++k) {
        const int kk = 4 * k + 2 * hi;
        af[k].x = xrow[kk];
        af[k].y = xrow[kk + 1];
    }

#pragma unroll
    for (int nt = 0; nt < MSG / 16; ++nt) {
        v8f acc = {};
        const int col = nt * 16 + row16;
#pragma unroll
        for (int k = 0; k < EMB / 4; ++k) {
            const int kk = 4 * k + 2 * hi;
            v2f bf;
            bf.x = W[kk * MSG + col];
            bf.y = W[(kk + 1) * MSG + col];
            acc = __builtin_amdgcn_wmma_f32_16x16x4_f32(
                false, af[k], false, bf, (short)0, acc, false, false);
        }
        const float bv = bias[col];
#pragma unroll
        for (int v = 0; v < 8; ++v) {
            const int r = tile * 16 + v + 8 * hi;   // C layout: row = v + 8*hi
            P[(size_t)r * MSG + col] = acc[v] + bv;
        }
    }
}

// ---------------------------------------------------------------------------
// Kernel B: fused edge stage. Per block: 128 edges.
//  Phase A: gate = E @ W_e via WMMA (each wave: 16 edges) -> LDS
//  Phase B: gather P_s/P_r, LayerNorm(32), SiLU, * gate, atomic scatter-add
// ---------------------------------------------------------------------------
__global__ __launch_bounds__(256) void edge_kernel(
    const float* __restrict__ Ps, const float* __restrict__ Pr,
    const float* __restrict__ E,  const int* __restrict__ senders,
    const int* __restrict__ receivers, const float* __restrict__ We,
    const float* __restrict__ ln_scale, const float* __restrict__ ln_bias,
    float* __restrict__ Macc)
{
    __shared__ float gate[128][MSG + 1];    // +1 pad: avoid bank conflicts

    const int lane      = threadIdx.x & 31;
    const int wave      = threadIdx.x >> 5;
    const int row16     = lane & 15;
    const int hi        = lane >> 4;
    const int ebase_blk = blockIdx.x * 128;
    const int ebase     = ebase_blk + wave * 16;

    // ---- Phase A: WMMA gate for this wave's 16 edges ----
    v2f af[MSG / 4];
    {
        const float* erow = E + (size_t)(ebase + row16) * MSG;
#pragma unroll
        for (int k = 0; k < MSG / 4; ++k) {
            const int kk = 4 * k + 2 * hi;
            af[k].x = erow[kk];
            af[k].y = erow[kk + 1];
        }
    }
#pragma unroll
    for (int nt = 0; nt < 2; ++nt) {
        v8f acc = {};
        const int col = nt * 16 + row16;
#pragma unroll
        for (int k = 0; k < MSG / 4; ++k) {
            const int kk = 4 * k + 2 * hi;
            v2f bf;
            bf.x = We[kk * MSG + col];
            bf.y = We[(kk + 1) * MSG + col];
            acc = __builtin_amdgcn_wmma_f32_16x16x4_f32(
                false, af[k], false, bf, (short)0, acc, false, false);
        }
#pragma unroll
        for (int v = 0; v < 8; ++v)
            gate[wave * 16 + v + 8 * hi][col] = acc[v];
    }
    __syncthreads();

    // ---- Phase B: two threads per edge, 16 columns each ----
    const int e_local = threadIdx.x >> 1;
    const int half    = threadIdx.x & 1;
    const int e       = ebase_blk + e_local;
    const int se      = senders[e];
    const int re      = receivers[e];
    const int col0    = half * 16;

    const float4* ps4 = (const float4*)(Ps + (size_t)se * MSG + col0);
    const float4* pr4 = (const float4*)(Pr + (size_t)re * MSG + col0);

    float x[16];
    float s1 = 0.0f, s2 = 0.0f;
#pragma unroll
    for (int i = 0; i < 4; ++i) {
        const float4 a = ps4[i];
        const float4 b = pr4[i];
        float t;
        t = a.x + b.x; x[4*i+0] = t; s1 += t; s2 += t * t;
        t = a.y + b.y; x[4*i+1] = t; s1 += t; s2 += t * t;
        t = a.z + b.z; x[4*i+2] = t; s1 += t; s2 += t * t;
        t = a.w + b.w; x[4*i+3] = t; s1 += t; s2 += t * t;
    }
    // combine the two halves of this edge (adjacent lanes, wave32)
    s1 += __shfl_xor(s1, 1, 32);
    s2 += __shfl_xor(s2, 1, 32);
    const float mean = s1 * (1.0f / MSG);
    const float var  = s2 * (1.0f / MSG) - mean * mean;
    const float rstd = rsqrtf(var + LN_EPS);

    float* mrow = Macc + (size_t)re * MSG + col0;
#pragma unroll
    for (int i = 0; i < 16; ++i) {
        const int c = col0 + i;
        float y = (x[i] - mean) * rstd * ln_scale[c] + ln_bias[c];
        y = silu_f(y);
        y *= gate[e_local][c];
        atomicAdd(&mrow[i], y);             // global_atomic_add_f32, L2-resident
    }
}

// ---------------------------------------------------------------------------
// Kernel C: Out = silu((Macc * norm[:,None]) @ W_out)   [N_NUC,32]@[32,128]
// ---------------------------------------------------------------------------
__global__ __launch_bounds__(256) void out_kernel(
    const float* __restrict__ Macc, const float* __restrict__ norm,
    const float* __restrict__ Wout, float* __restrict__ Out)
{
    const int lane  = threadIdx.x & 31;
    const int wave  = threadIdx.x >> 5;
    const int tile  = blockIdx.x * 8 + wave;
    const int NT    = N_NUC / 16;
    if (tile >= NT) return;
    const int row16 = lane & 15;
    const int hi    = lane >> 4;
    const int row   = tile * 16 + row16;
    const float nrm = norm[row];
    const float* mrow = Macc + (size_t)row * MSG;

    v2f af[MSG / 4];
#pragma unroll
    for (int k = 0; k < MSG / 4; ++k) {
        const int kk = 4 * k + 2 * hi;
        af[k].x = mrow[kk] * nrm;
        af[k].y = mrow[kk + 1] * nrm;
    }

#pragma unroll
    for (int nt = 0; nt < OUT / 16; ++nt) {
        v8f acc = {};
        const int col = nt * 16 + row16;
#pragma unroll
        for (int k = 0; k < MSG / 4; ++k) {
            const int kk = 4 * k + 2 * hi;
            v2f bf;
            bf.x = Wout[kk * OUT + col];
            bf.y = Wout[(kk + 1) * OUT + col];
            acc = __builtin_amdgcn_wmma_f32_16x16x4_f32(
                false, af[k], false, bf, (short)0, acc, false, false);
        }
#pragma unroll
        for (int v = 0; v < 8; ++v) {
            const int r = tile * 16 + v + 8 * hi;
            Out[(size_t)r * OUT + col] = silu_f(acc[v]);
        }
    }
}

// ---------------------------------------------------------------------------
extern "C" void kernel_launch(void* const* d_in, const int* in_sizes, int n_in,
                              void* d_out, int out_size, void* d_ws, size_t ws_size,
                              hipStream_t stream)
{
    const float* s_embed   = (const float*)d_in[0];
    const float* r_embed   = (const float*)d_in[1];
    const float* e_embed   = (const float*)d_in[2];
    const float* norm      = (const float*)d_in[3];
    const int*   senders   = (const int*)d_in[4];
    const int*   receivers = (const int*)d_in[5];
    const float* W_s       = (const float*)d_in[6];
    const float* b_s       = (const float*)d_in[7];
    const float* W_r       = (const float*)d_in[8];
    const float* b_r       = (const float*)d_in[9];
    const float* ln_scale  = (const float*)d_in[10];
    const float* ln_bias   = (const float*)d_in[11];
    const float* W_e       = (const float*)d_in[12];
    const float* W_out     = (const float*)d_in[13];
    float* out = (float*)d_out;

    // workspace: P_s | P_r | Macc  (12.8 MB each, 38.4 MB total, L2-resident)
    float* Ps   = (float*)d_ws;
    float* Pr   = Ps + (size_t)N_NUC * MSG;
    float* Macc = Pr + (size_t)N_NUC * MSG;

    hipMemsetAsync(Macc, 0, (size_t)N_NUC * MSG * sizeof(float), stream);

    const int ntiles  = N_NUC / 16;          // 6250
    const int blocksA = (ntiles + 7) / 8;    // 782 blocks x 8 waves
    node_proj_kernel<<<blocksA, 256, 0, stream>>>(s_embed, W_s, b_s, Ps);
    node_proj_kernel<<<blocksA, 256, 0, stream>>>(r_embed, W_r, b_r, Pr);

    const int blocksB = N_NN / 128;          // 12500
    edge_kernel<<<blocksB, 256, 0, stream>>>(Ps, Pr, e_embed, senders, receivers,
                                             W_e, ln_scale, ln_bias, Macc);

    out_kernel<<<blocksA, 256, 0, stream>>>(Macc, norm, W_out, out);
}